// ISTSimulator_2765958939456
// MI455X (gfx1250) — compile-verified
//
#include <hip/hip_runtime.h>

// ISTSimulator: 131072 independent 200-step scalar recurrences.
// One (b,v) sequence per lane (wave32). deltaR values are staged per-wave in
// LDS (pitch-33 tile, bank-conflict-free) and flushed with coalesced stores,
// synchronized with the gfx1250 split-counter s_wait_dscnt.

#define IST_EPS 1e-12f
constexpr int IST_L = 200;
constexpr int IST_P = 7;
constexpr int WAVES_PER_BLOCK = 8;

template <int C>
__device__ __forceinline__ void ist_chunk(
    int n0, float lam, float beta, float kd, float mu, float kf,
    float n_c, float inv_wt, float& D, float& F,
    float (*tile)[33],          // this wave's 32x33 LDS tile
    int lane, int waveSeqBase, int nseq,
    float* __restrict__ out)
{
    // ---- compute C steps of the recurrence, one deltaR per step into LDS ----
    #pragma unroll 4
    for (int k = 0; k < C; ++k) {
        float n = (float)(n0 + k);
        float omd = fmaxf(1.0f - D, IST_EPS);
        // D += lam * omd^beta  (exp(beta*log(omd))), clamp to 1-eps
        D = fminf(fmaf(lam, __expf(beta * __logf(omd)), D), 1.0f - IST_EPS);
        // g = sigmoid((n - n_c)/w_t)
        float x = (n - n_c) * inv_wt;
        float g = __builtin_amdgcn_rcpf(1.0f + __expf(-x));
        // F = clip((1 + mu*g)*F + eps*g, 0, 1-eps)
        F = fminf(fmaxf(fmaf(mu * g, F, F) + IST_EPS * g, 0.0f), 1.0f - IST_EPS);
        tile[lane][k] = fmaf(kd, D, kf * F);      // ds_store_b32
    }

    // Wait for this wave's LDS stores before the transposed read-back.
    // gfx1250 split dependency counter (replaces lgkmcnt).
    asm volatile("s_wait_dscnt 0x0" ::: "memory");

    // ---- coalesced flush: 32 lanes sweep the tile row-major ----
    // C==32: row = it, col = lane  -> one 128B contiguous store per iteration.
    // C==8 : 4 rows x 8 lanes per iteration (4 contiguous 32B segments).
    #pragma unroll 8
    for (int it = 0; it < C; ++it) {
        int flat = it * 32 + lane;
        int row  = flat / C;
        int col  = flat - row * C;
        int seq  = waveSeqBase + row;
        if (seq < nseq) {
            out[seq * IST_L + n0 + col] = tile[row][col];   // coalesced global_store_b32
        }
    }
    // Same-wave LDS ops complete in order, so the next chunk's ds_store after
    // these ds_loads (WAR on the tile) needs no extra wait.
}

__global__ __launch_bounds__(256)
void ist_sim_kernel(const float* __restrict__ params, float* __restrict__ out, int nseq)
{
    __shared__ float tile[WAVES_PER_BLOCK][32][33];   // 33 KB: pitch 33 -> conflict-free

    const int lane = threadIdx.x & 31;
    const int wave = threadIdx.x >> 5;
    const int seq  = blockIdx.x * blockDim.x + threadIdx.x;
    const int waveSeqBase = blockIdx.x * (int)blockDim.x + wave * 32;

    const bool active = seq < nseq;
    const float* p = params + (size_t)(active ? seq : 0) * IST_P;
    __builtin_prefetch(p, 0, 0);                      // global_prefetch_b8

    // Per-sequence parameters (hoisted transforms)
    float lam    = 0.001f * fmaxf(p[0], 0.0f);
    float beta   = fmaf(10.0f, fmaxf(p[1], 0.0f), 1.0f);
    float kd     = p[2];
    float mu     = p[3];
    float kf     = p[4];
    float n_c    = p[5];
    float inv_wt = 1.0f / p[6];

    float D = 0.0f;
    float F = IST_EPS;
    float (*myTile)[33] = tile[wave];

    int n0 = 0;
    #pragma unroll 1
    for (int c = 0; c < 6; ++c) {                     // 6 * 32 = 192 steps
        ist_chunk<32>(n0, lam, beta, kd, mu, kf, n_c, inv_wt, D, F,
                      myTile, lane, waveSeqBase, nseq, out);
        n0 += 32;
    }
    ist_chunk<8>(n0, lam, beta, kd, mu, kf, n_c, inv_wt, D, F,  // final 8 steps
                 myTile, lane, waveSeqBase, nseq, out);
}

extern "C" void kernel_launch(void* const* d_in, const int* in_sizes, int n_in,
                              void* d_out, int out_size, void* d_ws, size_t ws_size,
                              hipStream_t stream)
{
    const float* params = (const float*)d_in[0];      // (B, 14, 1) float32
    float* out = (float*)d_out;                       // (B, 400) float32
    const int nseq = in_sizes[0] / IST_P;             // B * NUM_VARS = 131072

    const int block = 256;
    const int grid  = (nseq + block - 1) / block;     // 512 blocks
    ist_sim_kernel<<<grid, block, 0, stream>>>(params, out, nseq);
}